// Pooling_Block_86517821212880
// MI455X (gfx1250) — compile-verified
//
#include <hip/hip_runtime.h>
#include <hip/hip_bf16.h>

// ---------------------------------------------------------------------------
// PointNet++ set abstraction: FPS -> ball query -> group+maxpool -> 1x1 conv
// (WMMA f32 16x16x4) -> BN -> LeakyReLU.  Sizes fixed per reference:
//   B=8, N=16384, Cin=128, Cout=256, S=NPOINT=1024, ns=NSAMPLE=32, r=0.2
// ---------------------------------------------------------------------------

typedef float v2f __attribute__((ext_vector_type(2)));
typedef float v8f __attribute__((ext_vector_type(8)));

#define B_    8
#define N_    16384
#define CIN_  128
#define COUT_ 256
#define S_    1024
#define NS_   32
#define R2_   0.04f
#define BS_   (B_ * S_)     // 8192 centers
#define PPT_  16            // points per thread in FPS (N/1024)

// ---------------------------------------------------------------------------
// Kernel 1: furthest point sampling. One 1024-thread workgroup per batch.
// Each thread keeps its 16 points + running min-dist in registers.
// ---------------------------------------------------------------------------
__global__ __launch_bounds__(1024) void fps_kernel(const float* __restrict__ xyz,
                                                   float* __restrict__ new_xyz)
{
    const int b   = blockIdx.x;
    const int tid = threadIdx.x;
    const int lane = tid & 31;
    const int wv   = tid >> 5;

    const float* base = xyz + (size_t)b * N_ * 3;

    float px[PPT_], py[PPT_], pz[PPT_], mind[PPT_];
#pragma unroll
    for (int i = 0; i < PPT_; ++i) {
        const int n = tid * PPT_ + i;
        px[i] = base[n * 3 + 0];
        py[i] = base[n * 3 + 1];
        pz[i] = base[n * 3 + 2];
        mind[i] = 1e10f;
    }

    __shared__ float sv[32];
    __shared__ int   si[32];
    __shared__ float scur[3];
    __shared__ int   sbi;

    // first selected point is index 0 (reference semantics)
    float cx = base[0], cy = base[1], cz = base[2];

    for (int s = 0; s < S_; ++s) {
        if (tid == 0) {
            float* o = new_xyz + ((size_t)b * S_ + s) * 3;
            o[0] = cx; o[1] = cy; o[2] = cz;
        }
        // update min-dist, local argmax (prefer lowest index on ties)
        float bv = -1.0f; int bi = 0;
#pragma unroll
        for (int i = 0; i < PPT_; ++i) {
            const float dx = px[i] - cx, dy = py[i] - cy, dz = pz[i] - cz;
            const float d = dx * dx + dy * dy + dz * dz;
            const float m = fminf(mind[i], d);
            mind[i] = m;
            if (m > bv) { bv = m; bi = tid * PPT_ + i; }
        }
        // wave32 argmax reduce
        for (int off = 16; off > 0; off >>= 1) {
            const float ov = __shfl_xor(bv, off, 32);
            const int   oi = __shfl_xor(bi, off, 32);
            if (ov > bv || (ov == bv && oi < bi)) { bv = ov; bi = oi; }
        }
        if (lane == 0) { sv[wv] = bv; si[wv] = bi; }
        __syncthreads();
        if (wv == 0) {
            float v = sv[lane]; int i2 = si[lane];
            for (int off = 16; off > 0; off >>= 1) {
                const float ov = __shfl_xor(v, off, 32);
                const int   oi = __shfl_xor(i2, off, 32);
                if (ov > v || (ov == v && oi < i2)) { v = ov; i2 = oi; }
            }
            if (lane == 0) sbi = i2;
        }
        __syncthreads();
        const int best = sbi;
        if ((best >> 4) == tid) {                  // owner thread broadcasts coords
            const int slot = best & (PPT_ - 1);
            float wx = 0.f, wy = 0.f, wz = 0.f;
#pragma unroll
            for (int i = 0; i < PPT_; ++i)
                if (i == slot) { wx = px[i]; wy = py[i]; wz = pz[i]; }
            scur[0] = wx; scur[1] = wy; scur[2] = wz;
        }
        __syncthreads();
        cx = scur[0]; cy = scur[1]; cz = scur[2];
    }
}

// ---------------------------------------------------------------------------
// Kernel 2: transpose feats [B,C,N] -> featsT [B,N,C] (coalesced both ways)
// ---------------------------------------------------------------------------
__global__ __launch_bounds__(256) void transpose_kernel(const float* __restrict__ feats,
                                                        float* __restrict__ featsT)
{
    __shared__ float tile[32][33];
    const int b  = blockIdx.z;
    const int c0 = blockIdx.y * 32;
    const int n0 = blockIdx.x * 32;
    const int tx = threadIdx.x, ty = threadIdx.y;
#pragma unroll
    for (int r = ty; r < 32; r += 8)
        tile[r][tx] = feats[(size_t)(b * CIN_ + c0 + r) * N_ + n0 + tx];
    __syncthreads();
#pragma unroll
    for (int r = ty; r < 32; r += 8)
        featsT[((size_t)b * N_ + n0 + r) * CIN_ + c0 + tx] = tile[tx][r];
}

// ---------------------------------------------------------------------------
// Kernel 3: fused ball query (first-32-in-order, early exit) + gather + maxpool.
// One wave per center.  Output pooled is channel-major: pooled[c][b*S+s].
// ---------------------------------------------------------------------------
__global__ __launch_bounds__(256) void ballpool_kernel(const float* __restrict__ xyz,
                                                       const float* __restrict__ new_xyz,
                                                       const float* __restrict__ featsT,
                                                       float* __restrict__ pooled)
{
    __shared__ int sidx[8][NS_];
    const int wv   = threadIdx.x >> 5;
    const int lane = threadIdx.x & 31;
    const int g    = blockIdx.x * 8 + wv;   // global center index, 0..8191
    const int b    = g >> 10;

    const float* base = xyz + (size_t)b * N_ * 3;
    const float* c    = new_xyz + (size_t)g * 3;
    const float cx = c[0], cy = c[1], cz = c[2];

    int total = 0;
    for (int n0 = 0; (n0 < N_) && (total < NS_); n0 += 32) {
        const int n = n0 + lane;
        const float dx = base[n * 3 + 0] - cx;
        const float dy = base[n * 3 + 1] - cy;
        const float dz = base[n * 3 + 2] - cz;
        const float d2 = dx * dx + dy * dy + dz * dz;
        const bool in = d2 < R2_;
        const unsigned m = __builtin_amdgcn_ballot_w32(in);
        const int pos = total + __popc(m & ((1u << lane) - 1u));
        if (in && pos < NS_) sidx[wv][pos] = n;
        total += __popc(m);
    }
    // fill empty slots with the first valid index (or 0 if none) -- ref semantics
    if (lane >= total) sidx[wv][lane] = (total > 0) ? sidx[wv][0] : 0;

    // gather 32 neighbor rows of 128 contiguous channels, running max.
    // lane owns channels 4*lane .. 4*lane+3 (float4 per row).
    float4 acc = make_float4(-3.4e38f, -3.4e38f, -3.4e38f, -3.4e38f);
    for (int j = 0; j < NS_; ++j) {
        const int nj = sidx[wv][j];
        const float4* row = reinterpret_cast<const float4*>(featsT + ((size_t)b * N_ + nj) * CIN_);
        const float4 v = row[lane];
        acc.x = fmaxf(acc.x, v.x);
        acc.y = fmaxf(acc.y, v.y);
        acc.z = fmaxf(acc.z, v.z);
        acc.w = fmaxf(acc.w, v.w);
    }
    const int cb = lane * 4;
    pooled[(size_t)(cb + 0) * BS_ + g] = acc.x;
    pooled[(size_t)(cb + 1) * BS_ + g] = acc.y;
    pooled[(size_t)(cb + 2) * BS_ + g] = acc.z;
    pooled[(size_t)(cb + 3) * BS_ + g] = acc.w;
}

// ---------------------------------------------------------------------------
// Kernel 4: y = W[256,128] @ pooled[128, 8192] via V_WMMA_F32_16X16X4_F32,
// fused BN (inference) + LeakyReLU(0.2).  Each wave computes a 16(M)x64(N)
// tile: 4 accumulators, K=128 -> 128 wmma per wave.  2048 waves total.
// A layout (16x4 f32): lane<16 holds M=lane, {K0,K1}; lane>=16 holds M=lane-16,
// {K2,K3}.  B layout (4x16 f32): lane<16 holds N=lane, {K0,K1}; lane>=16 {K2,K3}.
// C/D layout: VGPR r <-> M = r + 8*(lane>=16), N = lane&15.
// ---------------------------------------------------------------------------
__global__ __launch_bounds__(256) void gemm_bn_lrelu_kernel(const float* __restrict__ Wmat,
                                                            const float* __restrict__ pooled,
                                                            const float* __restrict__ gamma,
                                                            const float* __restrict__ beta,
                                                            const float* __restrict__ mean,
                                                            const float* __restrict__ var,
                                                            float* __restrict__ out)
{
    const int wid  = blockIdx.x * (blockDim.x >> 5) + (threadIdx.x >> 5);
    const int lane = threadIdx.x & 31;
    const int mo = wid & 15;   // 16 tiles over Cout=256
    const int nt = wid >> 4;   // 128 tiles of 64 centers over 8192
    const int hi = lane >> 4;
    const int lo = lane & 15;

    const float* Wrow   = Wmat + (size_t)(mo * 16 + lo) * CIN_;
    const int    colb   = nt * 64 + lo;

    v8f acc[4] = {v8f{}, v8f{}, v8f{}, v8f{}};

#pragma unroll
    for (int k0 = 0; k0 < CIN_; k0 += 4) {
        const int ka = k0 + 2 * hi;
        v2f a;
        a.x = Wrow[ka];
        a.y = Wrow[ka + 1];

        const float* p = pooled + (size_t)ka * BS_ + colb;
        v2f b0, b1, b2, b3;
        b0.x = p[0];        b0.y = p[BS_];
        b1.x = p[16];       b1.y = p[BS_ + 16];
        b2.x = p[32];       b2.y = p[BS_ + 32];
        b3.x = p[48];       b3.y = p[BS_ + 48];

        acc[0] = __builtin_amdgcn_wmma_f32_16x16x4_f32(false, a, false, b0, (short)0, acc[0], false, false);
        acc[1] = __builtin_amdgcn_wmma_f32_16x16x4_f32(false, a, false, b1, (short)0, acc[1], false, false);
        acc[2] = __builtin_amdgcn_wmma_f32_16x16x4_f32(false, a, false, b2, (short)0, acc[2], false, false);
        acc[3] = __builtin_amdgcn_wmma_f32_16x16x4_f32(false, a, false, b3, (short)0, acc[3], false, false);
    }

    // fold BN: y = v*sc + sh,  sc = gamma/sqrt(var+eps), sh = beta - mean*sc
    float sc[8], sh[8];
#pragma unroll
    for (int r = 0; r < 8; ++r) {
        const int o = mo * 16 + r + 8 * hi;
        const float s = gamma[o] * rsqrtf(var[o] + 1e-5f);
        sc[r] = s;
        sh[r] = beta[o] - mean[o] * s;
    }

#pragma unroll
    for (int ns = 0; ns < 4; ++ns) {
        const int col = nt * 64 + ns * 16 + lo;   // global center index
        const int bb  = col >> 10;
        const int s   = col & 1023;
#pragma unroll
        for (int r = 0; r < 8; ++r) {
            const int o = mo * 16 + r + 8 * hi;
            float v = acc[ns][r] * sc[r] + sh[r];
            v = (v >= 0.0f) ? v : 0.2f * v;
            out[((size_t)(bb * COUT_ + o)) * S_ + s] = v;
        }
    }
}

// ---------------------------------------------------------------------------
extern "C" void kernel_launch(void* const* d_in, const int* in_sizes, int n_in,
                              void* d_out, int out_size, void* d_ws, size_t ws_size,
                              hipStream_t stream)
{
    const float* xyz   = (const float*)d_in[0];
    const float* feats = (const float*)d_in[1];
    const float* Wmat  = (const float*)d_in[2];
    const float* gamma = (const float*)d_in[3];
    const float* beta  = (const float*)d_in[4];
    const float* mean  = (const float*)d_in[5];
    const float* var   = (const float*)d_in[6];

    char* ws = (char*)d_ws;
    const size_t featsT_bytes = (size_t)B_ * N_ * CIN_ * sizeof(float);   // 64 MB
    const size_t pooled_bytes = (size_t)CIN_ * BS_ * sizeof(float);       // 4 MB
    float* featsT = (float*)ws;
    float* pooled = (float*)(ws + featsT_bytes);
    float* nxyz   = (float*)(ws + featsT_bytes + pooled_bytes);           // 96 KB

    // 1) FPS: one workgroup per batch
    fps_kernel<<<B_, 1024, 0, stream>>>(xyz, nxyz);

    // 2) transpose feats -> [B,N,C] for contiguous neighbor rows
    transpose_kernel<<<dim3(N_ / 32, CIN_ / 32, B_), dim3(32, 8, 1), 0, stream>>>(feats, featsT);

    // 3) fused ball query + gather + maxpool, one wave per center
    ballpool_kernel<<<BS_ / 8, 256, 0, stream>>>(xyz, nxyz, featsT, pooled);

    // 4) WMMA fp32 GEMM + BN + LeakyReLU
    gemm_bn_lrelu_kernel<<<256, 256, 0, stream>>>(Wmat, pooled, gamma, beta, mean, var, (float*)d_out);
}